// MakeCutouts_29729763623593
// MI455X (gfx1250) — compile-verified
//
#include <hip/hip_runtime.h>
#include <stdint.h>

// Problem constants (match reference)
#define CUTN 32
#define BB   2
#define CC   3
#define HH   512
#define WW   512
#define CS   224

#define ROWS_PER_BLK 8
#define IBLKS (CS / ROWS_PER_BLK)          /* 28 stripes of 8 output rows   */
#define MAX_ROWS 20                         /* max input rows per stripe     */
#define THREADS 256

typedef int v4i __attribute__((ext_vector_type(4)));
typedef v4i __attribute__((address_space(1)))* gptr_v4i;   // global
typedef v4i __attribute__((address_space(3)))* lptr_v4i;   // LDS

#if defined(__HIP_DEVICE_COMPILE__)
#if __has_builtin(__builtin_amdgcn_global_load_async_to_lds_b128)
#define HAVE_ASYNC_B128 1
#endif
#endif

__global__ __launch_bounds__(THREADS) void
MakeCutouts_kernel(const float* __restrict__ input,
                   const int* __restrict__ sizes_,
                   const int* __restrict__ offx_,
                   const int* __restrict__ offy_,
                   float* __restrict__ out)
{
    __shared__ float tile[MAX_ROWS * WW];   // 40 KB -> 8 blocks/WGP

    const int bid = blockIdx.x;
    const int ib  = bid % IBLKS;                 // output-row stripe
    const int bc  = (bid / IBLKS) % (BB * CC);   // b*C + c
    const int n   = bid / (IBLKS * BB * CC);     // cutout index
    const int tid = threadIdx.x;

    const int size = sizes_[n];
    const int offx = offx_[n];
    const int offy = offy_[n];

    const int i0  = ib * ROWS_PER_BLK;
    const int sy0 = offy + (int)((unsigned)(i0 * size) / CS);
    const int eyN = offy + (int)(((unsigned)((i0 + ROWS_PER_BLK) * size) + CS - 1) / CS);
    const int nrows = eyN - sy0;                 // <= 20

    const float* src = input + ((size_t)bc * HH + (size_t)sy0) * WW;

    // ---- Stage nrows x 512 fp32 rows into LDS (16B chunks, 128/row) ----
    const int chunks = nrows << 7;               // nrows * 128
    for (int t = tid; t < chunks; t += THREADS) {
        const int row = t >> 7;
        const int ch  = t & 127;
        const float* g = src + row * WW + (ch << 2);
        float*       l = &tile[row * WW + (ch << 2)];
#ifdef HAVE_ASYNC_B128
        __builtin_amdgcn_global_load_async_to_lds_b128(
            (gptr_v4i)g, (lptr_v4i)l, 0, 0);
#else
        *(float4*)l = *(const float4*)g;         // global_load_b128 + ds_store_b128
#endif
    }
#ifdef HAVE_ASYNC_B128
#if __has_builtin(__builtin_amdgcn_s_wait_asynccnt)
    __builtin_amdgcn_s_wait_asynccnt(0);
#else
    asm volatile("s_wait_asynccnt 0" ::: "memory");
#endif
#endif
    __syncthreads();

    // ---- Each thread produces 7 output pixels of the 8x224 stripe ----
    float* obase = out + ((size_t)n * (BB * CC) + (size_t)bc) * (CS * CS);
    for (int p = tid; p < ROWS_PER_BLK * CS; p += THREADS) {
        const int il = p / CS;
        const int j  = p - il * CS;
        const int i  = i0 + il;

        // PyTorch adaptive-avg-pool bin bounds, shifted by crop offset
        const int sy = offy + (int)((unsigned)(i * size) / CS);
        const int ey = offy + (int)(((unsigned)((i + 1) * size) + CS - 1) / CS);
        const int sx = offx + (int)((unsigned)(j * size) / CS);
        const int ex = offx + (int)(((unsigned)((j + 1) * size) + CS - 1) / CS);

        float acc = 0.0f;
        for (int y = sy; y < ey; ++y) {          // <= 3 iterations
            const float* r = &tile[(y - sy0) * WW];
            for (int x = sx; x < ex; ++x)        // <= 3 iterations
                acc += r[x];
        }
        const float inv = 1.0f / (float)((ey - sy) * (ex - sx));
        // Output is write-once streaming data: non-temporal store (TH=NT)
        // keeps L2 free for the re-read input tiles.
        __builtin_nontemporal_store(acc * inv, &obase[(size_t)i * CS + j]);
    }
}

extern "C" void kernel_launch(void* const* d_in, const int* in_sizes, int n_in,
                              void* d_out, int out_size, void* d_ws, size_t ws_size,
                              hipStream_t stream) {
    const float* input = (const float*)d_in[0];
    const int*   sizes = (const int*)d_in[1];
    const int*   offx  = (const int*)d_in[2];
    const int*   offy  = (const int*)d_in[3];
    float*       out   = (float*)d_out;

    const dim3 grid(CUTN * BB * CC * IBLKS);     // 5376 blocks
    MakeCutouts_kernel<<<grid, THREADS, 0, stream>>>(input, sizes, offx, offy, out);
}